// SCVC_22986664968204
// MI455X (gfx1250) — compile-verified
//
#include <hip/hip_runtime.h>

typedef float v2f __attribute__((ext_vector_type(2)));
typedef float v8f __attribute__((ext_vector_type(8)));

#define NB   32
#define IN1  4
#define IN2  4
#define OUTC 128
#define DX   128
#define DY   128

// ---------------------------------------------------------------------------
// Kernel 0: collapse tiled-channel conv weights:
//   war[o][c][tap] = sum_{t4} W_A[o][t4*4+c][tap]   (xA channel t4*4+c reads x[c])
// ---------------------------------------------------------------------------
__global__ void SCVC_wred_kernel(const float* __restrict__ W_A,
                                 const float* __restrict__ W_B,
                                 float* __restrict__ war,
                                 float* __restrict__ wbr) {
    int idx = blockIdx.x * blockDim.x + threadIdx.x;   // [0, 128*4*2)
    if (idx >= OUTC * IN1 * 2) return;
    int tap = idx & 1;
    int c   = (idx >> 1) & 3;
    int o   = idx >> 3;
    float sa = 0.f, sb = 0.f;
#pragma unroll
    for (int t4 = 0; t4 < IN2; ++t4) {
        sa += W_A[(o * 16 + t4 * 4 + c) * 2 + tap];
        sb += W_B[(o * 16 + t4 * 4 + c) * 2 + tap];
    }
    war[idx] = sa;
    wbr[idx] = sb;
}

// ---------------------------------------------------------------------------
// Kernel 1: tiny dilated convs (kernel 2, dilation 2, pad 1):
//   s[n,o,i] = bias[o] + sum_c war[o,c,0]*x[n,c,i-1] + war[o,c,1]*x[n,c,i+1]
//   t[n,o,j] =            sum_c wbr[o,c,0]*y[n,c,j-1] + wbr[o,c,1]*y[n,c,j+1]
// ---------------------------------------------------------------------------
__global__ void SCVC_conv_kernel(const float* __restrict__ x,
                                 const float* __restrict__ y,
                                 const float* __restrict__ bias,
                                 const float* __restrict__ war,
                                 const float* __restrict__ wbr,
                                 float* __restrict__ s,
                                 float* __restrict__ t) {
    int idx = blockIdx.x * blockDim.x + threadIdx.x;   // [0, 2*32*128*128)
    const int HALF = NB * OUTC * DX;
    int which = (idx >= HALF) ? 1 : 0;
    int r = which ? (idx - HALF) : idx;
    int i = r & 127;
    int o = (r >> 7) & 127;
    int n = r >> 14;
    const float* inp = which ? y : x;
    const float* w   = which ? wbr : war;
    float acc = which ? 0.f : bias[o];
#pragma unroll
    for (int c = 0; c < 4; ++c) {
        float vm = (i >= 1)   ? inp[(n * 4 + c) * 128 + i - 1] : 0.f;
        float vp = (i <= 126) ? inp[(n * 4 + c) * 128 + i + 1] : 0.f;
        acc += w[(o * 4 + c) * 2 + 0] * vm + w[(o * 4 + c) * 2 + 1] * vp;
    }
    (which ? t : s)[r] = acc;
}

// ---------------------------------------------------------------------------
// Kernel 2: store-bound WMMA kernel. One wave32 per (n, i, o-block of 16).
// D[o,j] = sum_k C[o,k] * (x[n,c1,i]*y[n,c2,j]) via 4 chained
// V_WMMA_F32_16X16X4_F32; accumulator pre-loaded with s[n,o,i] + t[n,o,j].
//
// Layouts (CDNA5 ISA, 32-bit, wave32):
//   A 16x4 : lanes 0-15 -> M rows; VGPR0/1 = K{0,1} (lanes<16), K{2,3} (lanes>=16)
//   B  4x16: lanes hold N columns; reg0/1 = K rows {0,2}/{1,3} mirrored by half
//   C/D    : reg r, lanes 0-15 -> M=r, N=lane; lanes 16-31 -> M=r+8
// ---------------------------------------------------------------------------
__global__ void __launch_bounds__(256) SCVC_bcp_kernel(
    const float* __restrict__ x, const float* __restrict__ y,
    const float* __restrict__ C, const float* __restrict__ s,
    const float* __restrict__ t, float* __restrict__ out) {
    const int lane  = threadIdx.x & 31;
    const int hf    = lane >> 4;     // lane half: selects K rows / M+8
    const int lid   = lane & 15;
    const int wid   = (blockIdx.x * blockDim.x + threadIdx.x) >> 5;
    const int oblk  = wid & 7;
    const int i     = (wid >> 3) & 127;
    const int n     = wid >> 10;
    const int obase = oblk * 16;

    // A tiles: constant per wave. Chunk kk covers k = 4*kk + {0..3} (c1=kk, c2=0..3)
    v2f A[4];
    const int o_l = obase + lid;
#pragma unroll
    for (int kk = 0; kk < 4; ++kk) {
        A[kk].x = C[o_l * 16 + kk * 4 + 2 * hf + 0];
        A[kk].y = C[o_l * 16 + kk * 4 + 2 * hf + 1];
    }
    // x scalars for this (n, i)
    float xk[4];
#pragma unroll
    for (int c = 0; c < 4; ++c) xk[c] = x[(n * 4 + c) * 128 + i];
    // s[n, o, i] per accumulator register (o = obase + r + 8*hf)
    float sv[8];
#pragma unroll
    for (int r = 0; r < 8; ++r)
        sv[r] = s[(n * 128 + obase + r + 8 * hf) * 128 + i];

    for (int jb = 0; jb < 8; ++jb) {
        const int j = jb * 16 + lid;
        float yv[4];
#pragma unroll
        for (int c = 0; c < 4; ++c) yv[c] = y[(n * 4 + c) * 128 + j];
        const float ylo = hf ? yv[2] : yv[0];   // B reg0 rows: c2 = 0 (lo) / 2 (hi)
        const float yhi = hf ? yv[3] : yv[1];   // B reg1 rows: c2 = 1 (lo) / 3 (hi)

        v8f acc;
#pragma unroll
        for (int r = 0; r < 8; ++r)
            acc[r] = sv[r] + t[(n * 128 + obase + r + 8 * hf) * 128 + j];

#pragma unroll
        for (int kk = 0; kk < 4; ++kk) {
            v2f B;
            B.x = xk[kk] * ylo;
            B.y = xk[kk] * yhi;
            // (neg_a, A, neg_b, B, c_mod, C, reuse_a, reuse_b)
            acc = __builtin_amdgcn_wmma_f32_16x16x4_f32(
                false, A[kk], false, B, (short)0, acc, false, false);
        }

        // store: reg r -> out[n, obase + r + 8*hf, i, jb*16 + lid]
        // half-wave writes 64 contiguous bytes per register (full segments)
#pragma unroll
        for (int r = 0; r < 8; ++r)
            out[((n * 128 + obase + r + 8 * hf) * 128 + i) * 128 + j] = acc[r];
    }
}

// ---------------------------------------------------------------------------
extern "C" void kernel_launch(void* const* d_in, const int* in_sizes, int n_in,
                              void* d_out, int out_size, void* d_ws, size_t ws_size,
                              hipStream_t stream) {
    const float* x    = (const float*)d_in[0];   // [32,4,128]
    const float* y    = (const float*)d_in[1];   // [32,4,128]
    const float* C    = (const float*)d_in[2];   // [128,16]
    const float* W_A  = (const float*)d_in[3];   // [128,16,2]
    const float* W_B  = (const float*)d_in[4];   // [128,16,2]
    const float* bias = (const float*)d_in[5];   // [128]
    float* out = (float*)d_out;                  // [32,128,128,128]

    float* ws  = (float*)d_ws;
    float* s   = ws;                       // 32*128*128 = 524288 floats (2 MB)
    float* t   = ws + 524288;              // 524288 floats (2 MB)
    float* war = ws + 1048576;             // 1024 floats
    float* wbr = ws + 1049600;             // 1024 floats

    // 0) reduce conv weights over tiled channels
    SCVC_wred_kernel<<<4, 256, 0, stream>>>(W_A, W_B, war, wbr);
    // 1) per-axis conv terms (+bias folded into s)
    SCVC_conv_kernel<<<4096, 256, 0, stream>>>(x, y, bias, war, wbr, s, t);
    // 2) main WMMA kernel: 32768 full waves, no divergence (EXEC all-ones)
    SCVC_bcp_kernel<<<4096, 256, 0, stream>>>(x, y, C, s, t, out);
}